// CapsuleNet_74285754351713
// MI455X (gfx1250) — compile-verified
//
#include <hip/hip_runtime.h>
#include <hip/hip_bf16.h>

typedef float v2f __attribute__((ext_vector_type(2)));
typedef float v8f __attribute__((ext_vector_type(8)));

#define BSZ   128
#define NLEN  100
#define DEMB  400
#define C1CH  256
#define H1    98
#define H2    93
#define K1    1200   // 3*400
#define K2    1536   // 6*256
#define NRC   19
#define INCAPS 2976
#define CAPSK 23808  // 2976*8

__device__ __forceinline__ v8f wmma4(v2f a, v2f b, v8f c) {
    // D = A(16x4) * B(4x16) + C(16x16), fp32
    return __builtin_amdgcn_wmma_f32_16x16x4_f32(false, a, false, b, (short)0, c, false, false);
}

// ---------------- embedding gather: xc[B,100,400] ----------------
__global__ __launch_bounds__(128) void embed_kernel(
    const int* __restrict__ x, const int* __restrict__ dist1, const int* __restrict__ dist2,
    const float* __restrict__ pad_emb, const float* __restrict__ other_emb,
    const float* __restrict__ pad_pos, const float* __restrict__ other_pos,
    float* __restrict__ xc) {
    int tok = blockIdx.x;                  // B*N tokens
    int xi = x[tok], d1 = dist1[tok], d2 = dist2[tok];
    float* dst = xc + (long long)tok * DEMB;
    for (int t = threadIdx.x; t < DEMB; t += 128) {
        float v;
        if (t < 300)      v = (xi == 0)  ? pad_emb[t]        : other_emb[(xi - 1) * 300 + t];
        else if (t < 350) v = (d1 == 199) ? pad_pos[t - 300] : other_pos[d1 * 50 + (t - 300)];
        else              v = (d2 == 199) ? pad_pos[t - 350] : other_pos[d2 * 50 + (t - 350)];
        dst[t] = v;
    }
}

// ---------------- conv1 as WMMA GEMM: y1[B*98, 256], relu(bias+...) ----------------
__global__ __launch_bounds__(128) void conv1_kernel(
    const float* __restrict__ xc, const float* __restrict__ w1,
    const float* __restrict__ bias, float* __restrict__ y1) {
    int wid = threadIdx.x >> 5, lane = threadIdx.x & 31;
    int tile = blockIdx.x * 4 + wid;       // 784 mtiles x 16 ntiles
    int nt = tile & 15, mt = tile >> 4;
    int half = lane >> 4, l16 = lane & 15;
    int m = mt * 16 + l16;
    int b = m / H1, h = m % H1;
    const float* arow = xc + b * (NLEN * DEMB) + h * DEMB;   // contiguous 1200-wide window
    int n = nt * 16 + l16;
    const float* brow = w1 + n * K1;                         // B stored [N][K]
    int kb = half * 2;
    v8f acc = {};
    for (int k0 = 0; k0 < K1; k0 += 4) {
        v2f a = *(const v2f*)(arow + k0 + kb);
        v2f bb = *(const v2f*)(brow + k0 + kb);
        acc = wmma4(a, bb, acc);
    }
    float bs = bias[n];
    int mb = mt * 16 + half * 8;
#pragma unroll
    for (int j = 0; j < 8; ++j) {
        float v = acc[j] + bs;
        y1[(mb + j) * C1CH + n] = v > 0.f ? v : 0.f;
    }
}

// ---------------- prim_w transpose: wt[c][kh*256+cin] ----------------
__global__ void transpose_primw_kernel(const float* __restrict__ pw, float* __restrict__ wt) {
    int t = blockIdx.x * 256 + threadIdx.x;     // 256*256*6
    if (t >= C1CH * C1CH * 6) return;
    int kh = t % 6; int r = t / 6; int cin = r % C1CH; int c = r / C1CH;
    wt[c * K2 + kh * C1CH + cin] = pw[t];
}

// ---------------- conv2 as WMMA GEMM -> y2f[b][c*93+h2] (+bias) ----------------
__global__ __launch_bounds__(128) void conv2_kernel(
    const float* __restrict__ y1, const float* __restrict__ wt,
    const float* __restrict__ bias, float* __restrict__ y2f) {
    int wid = threadIdx.x >> 5, lane = threadIdx.x & 31;
    int tile = blockIdx.x * 4 + wid;       // 744 mtiles x 16 ntiles
    int nt = tile & 15, mt = tile >> 4;
    int half = lane >> 4, l16 = lane & 15;
    int m = mt * 16 + l16;
    int b = m / H2, h2 = m % H2;
    const float* arow = y1 + b * (H1 * C1CH) + h2 * C1CH;    // contiguous 1536-wide window
    int n = nt * 16 + l16;
    const float* brow = wt + n * K2;
    int kb = half * 2;
    v8f acc = {};
    for (int k0 = 0; k0 < K2; k0 += 4) {
        v2f a = *(const v2f*)(arow + k0 + kb);
        v2f bb = *(const v2f*)(brow + k0 + kb);
        acc = wmma4(a, bb, acc);
    }
    float bs = bias[n];
    int mb = mt * 16 + half * 8;
#pragma unroll
    for (int j = 0; j < 8; ++j) {
        int mr = mb + j;
        int bj = mr / H2, hj = mr % H2;
        y2f[bj * CAPSK + n * H2 + hj] = acc[j] + bs;
    }
}

// ---------------- squash primary capsules ----------------
__global__ void squash_caps_kernel(const float* __restrict__ y2f, float* __restrict__ caps) {
    int t = blockIdx.x * 256 + threadIdx.x;     // B*2976 capsules
    if (t >= BSZ * INCAPS) return;
    const float* p = y2f + t * 8;
    float v[8], nn = 0.f;
#pragma unroll
    for (int d = 0; d < 8; ++d) { v[d] = p[d]; nn += v[d] * v[d]; }
    float norm = sqrtf(nn);
    float scale = (nn / (1.f + nn)) / (norm + 1e-8f);
    float* q = caps + t * 8;
#pragma unroll
    for (int d = 0; d < 8; ++d) q[d] = v[d] * scale;
}

// ---------------- s[b,n,o] = sum_i c[b,n,i] * x_hat[b,n,i,o] (fused GEMM, K=23808) ----
__global__ __launch_bounds__(128) void routing_sum_kernel(
    const float* __restrict__ caps, const float* __restrict__ rw,
    const float* __restrict__ cmat, float* __restrict__ s, int useC) {
    int wid = threadIdx.x >> 5, lane = threadIdx.x & 31;
    int wg = blockIdx.x * 4 + wid;         // 19 classes x 8 batch-tiles = 152 waves
    int n = wg >> 3, bt = wg & 7;
    int half = lane >> 4, l16 = lane & 15;
    int b = bt * 16 + l16;
    const float* arow = caps + b * CAPSK;
    const float* Bp = rw + n * (INCAPS * 128);
    const float* crow = cmat ? (cmat + (b * NRC + n) * INCAPS) : nullptr;
    int o = l16, kb = half * 2;
    v8f acc = {};
    for (int k0 = 0; k0 < CAPSK; k0 += 4) {
        int i = k0 >> 3;
        int d = (k0 & 7) + kb;
        float cv = useC ? crow[i] : (1.0f / 19.0f);
        v2f a = *(const v2f*)(arow + k0 + kb);
        a *= cv;
        v2f bb = *(const v2f*)(Bp + i * 128 + o * 8 + d);
        acc = wmma4(a, bb, acc);
    }
    int mb = bt * 16 + half * 8;
#pragma unroll
    for (int j = 0; j < 8; ++j)
        s[((mb + j) * NRC + n) * 16 + o] = acc[j];
}

// ---------------- squash(s) -> out, or final class norms -> d_out ----------------
__global__ void squash_out_kernel(const float* __restrict__ s, float* __restrict__ out,
                                  float* __restrict__ fin, int final_pass) {
    int t = blockIdx.x * 256 + threadIdx.x;     // B*19
    if (t >= BSZ * NRC) return;
    const float* p = s + t * 16;
    float v[16], nn = 0.f;
#pragma unroll
    for (int o = 0; o < 16; ++o) { v[o] = p[o]; nn += v[o] * v[o]; }
    float norm = sqrtf(nn);
    float scale = (nn / (1.f + nn)) / (norm + 1e-8f);
    if (final_pass) {
        fin[t] = scale * norm;                  // || squash(s) ||
    } else {
        float* q = out + t * 16;
#pragma unroll
        for (int o = 0; o < 16; ++o) q[o] = v[o] * scale;
    }
}

// ---------------- blog[b,n,i] (+)= sum_o out[b,n,o] * x_hat[b,n,i,o] ----------------
// recompute 16x16 x_hat tiles with 2 WMMAs, then shfl-reduce over o
__global__ __launch_bounds__(128) void blogit_kernel(
    const float* __restrict__ caps, const float* __restrict__ rw,
    const float* __restrict__ out, float* __restrict__ blog, int accumulate) {
    int wid = threadIdx.x >> 5, lane = threadIdx.x & 31;
    int wg = blockIdx.x * 4 + wid;              // 31 i-chunks x 19 x 8 = 4712 waves
    int ic = wg / 152; int rem = wg % 152;
    int n = rem >> 3, bt = rem & 7;
    int half = lane >> 4, l16 = lane & 15;
    int b = bt * 16 + l16;
    int o = l16, kb = half * 2;
    int mb = bt * 16 + half * 8;
    float ov[8];
#pragma unroll
    for (int j = 0; j < 8; ++j) ov[j] = out[((mb + j) * NRC + n) * 16 + o];
    for (int i = ic * 96; i < ic * 96 + 96; ++i) {
        const float* Bp = rw + (n * INCAPS + i) * 128;
        const float* ar = caps + (b * INCAPS + i) * 8;
        v8f acc = {};
        acc = wmma4(*(const v2f*)(ar + kb),     *(const v2f*)(Bp + o * 8 + kb),     acc);
        acc = wmma4(*(const v2f*)(ar + 4 + kb), *(const v2f*)(Bp + o * 8 + 4 + kb), acc);
        float part[8];
#pragma unroll
        for (int j = 0; j < 8; ++j) part[j] = acc[j] * ov[j];
#pragma unroll
        for (int mask = 1; mask < 16; mask <<= 1)
#pragma unroll
            for (int j = 0; j < 8; ++j) part[j] += __shfl_xor(part[j], mask, 32);
        if (l16 == 0) {
#pragma unroll
            for (int j = 0; j < 8; ++j) {
                int idx = ((mb + j) * NRC + n) * INCAPS + i;
                blog[idx] = accumulate ? (blog[idx] + part[j]) : part[j];
            }
        }
    }
}

// ---------------- softmax over classes (axis n) ----------------
__global__ void softmax_kernel(const float* __restrict__ blog, float* __restrict__ c) {
    int t = blockIdx.x * 256 + threadIdx.x;     // B*2976
    if (t >= BSZ * INCAPS) return;
    int b = t / INCAPS, i = t % INCAPS;
    const float* p = blog + (b * NRC) * INCAPS + i;
    float mx = -1e30f;
#pragma unroll
    for (int n = 0; n < NRC; ++n) mx = fmaxf(mx, p[n * INCAPS]);
    float e[NRC], sum = 0.f;
#pragma unroll
    for (int n = 0; n < NRC; ++n) { e[n] = expf(p[n * INCAPS] - mx); sum += e[n]; }
    float inv = 1.0f / sum;
    float* q = c + (b * NRC) * INCAPS + i;
#pragma unroll
    for (int n = 0; n < NRC; ++n) q[n * INCAPS] = e[n] * inv;
}

extern "C" void kernel_launch(void* const* d_in, const int* in_sizes, int n_in,
                              void* d_out, int out_size, void* d_ws, size_t ws_size,
                              hipStream_t stream) {
    const int*   x       = (const int*)d_in[0];
    const int*   dist1   = (const int*)d_in[3];
    const int*   dist2   = (const int*)d_in[4];
    const float* pad_emb = (const float*)d_in[5];
    const float* oth_emb = (const float*)d_in[6];
    const float* pad_pos = (const float*)d_in[7];
    const float* oth_pos = (const float*)d_in[8];
    const float* w1      = (const float*)d_in[9];
    const float* b1      = (const float*)d_in[10];
    const float* pw      = (const float*)d_in[11];
    const float* pb      = (const float*)d_in[12];
    const float* rw      = (const float*)d_in[13];

    float* ws   = (float*)d_ws;
    float* caps = ws + 0;            // 3,047,424
    float* blog = ws + 3047424;      // 7,237,632
    float* cmat = ws + 10285056;     // 7,237,632
    float* sbuf = ws + 17522688;     //    38,912
    float* obuf = ws + 17561600;     //    38,912
    float* xc   = ws + 17600512;     // 5,120,000 (dead after conv1)
    float* y2f  = xc;                // 3,047,424 (reuses xc region)
    float* y1   = ws + 22720512;     // 3,211,264
    float* wt   = ws + 25931776;     //   393,216

    embed_kernel<<<BSZ * NLEN, 128, 0, stream>>>(x, dist1, dist2, pad_emb, oth_emb, pad_pos, oth_pos, xc);
    conv1_kernel<<<3136, 128, 0, stream>>>(xc, w1, b1, y1);                 // 784*16 tiles / 4 waves
    transpose_primw_kernel<<<1536, 256, 0, stream>>>(pw, wt);
    conv2_kernel<<<2976, 128, 0, stream>>>(y1, wt, pb, y2f);                // 744*16 tiles / 4 waves
    squash_caps_kernel<<<1488, 256, 0, stream>>>(y2f, caps);

    // routing iteration 0 (c uniform = 1/19)
    routing_sum_kernel<<<38, 128, 0, stream>>>(caps, rw, nullptr, sbuf, 0);
    squash_out_kernel<<<10, 256, 0, stream>>>(sbuf, obuf, nullptr, 0);
    blogit_kernel<<<1178, 128, 0, stream>>>(caps, rw, obuf, blog, 0);
    softmax_kernel<<<1488, 256, 0, stream>>>(blog, cmat);
    // routing iteration 1
    routing_sum_kernel<<<38, 128, 0, stream>>>(caps, rw, cmat, sbuf, 1);
    squash_out_kernel<<<10, 256, 0, stream>>>(sbuf, obuf, nullptr, 0);
    blogit_kernel<<<1178, 128, 0, stream>>>(caps, rw, obuf, blog, 1);
    softmax_kernel<<<1488, 256, 0, stream>>>(blog, cmat);
    // routing iteration 2 -> final class norms
    routing_sum_kernel<<<38, 128, 0, stream>>>(caps, rw, cmat, sbuf, 1);
    squash_out_kernel<<<10, 256, 0, stream>>>(sbuf, nullptr, (float*)d_out, 1);
}